// MultiheadLocalAttention_33354716021311
// MI455X (gfx1250) — compile-verified
//
#include <hip/hip_runtime.h>
#include <hip/hip_bf16.h>
#include <cstdint>

#define D_MODEL 256
#define NHEAD   8
#define HID     32
#define MAX_DIS 7
#define WSZ     15
#define W2      225
#define IMG_H   48
#define IMG_W   48
#define NPIX    2304   // 48*48

// attention tiling
#define TQX 16
#define TQY 8
#define MAXTDX (TQX + 2 * MAX_DIS)            // 30
#define MAXTDY (TQY + 2 * MAX_DIS)            // 22
#define KTILE_FLOATS (MAXTDX * MAXTDY * HID)  // 21120 floats = 84,480 B per matrix

typedef __attribute__((ext_vector_type(2))) float v2f;
typedef __attribute__((ext_vector_type(8))) float v8f;
typedef __attribute__((ext_vector_type(4))) unsigned int v4u;
typedef __attribute__((ext_vector_type(8))) int v8i;
typedef __attribute__((ext_vector_type(4))) int v4i;

// ---------------------------------------------------------------------------
// WMMA fp32 fragment layouts (wave32, V_WMMA_F32_16X16X4_F32):
//  A (16x4):  lane L: m=L&15, a.x = A[m][k + 2*(L>>4)], a.y = A[m][k + 2*(L>>4)+1]
//  B (4x16):  lane L: n=L&15, b.x = B[k + 2*(L>>4)][n], b.y = B[k + 2*(L>>4)+1][n]
//  C (16x16): vgpr r, lane L: row = r + 8*(L>>4), col = L&15
// ---------------------------------------------------------------------------

// Projection GEMMs: Y[o][p] = sum_c W[o][c] * X[c][p] + b[o]   (M=256,N=2304,K=256)
__global__ __launch_bounds__(32) void proj3_gemm(
    const float* __restrict__ q,  const float* __restrict__ k,  const float* __restrict__ v,
    const float* __restrict__ Wq, const float* __restrict__ Wk, const float* __restrict__ Wv,
    const float* __restrict__ bq, const float* __restrict__ bk, const float* __restrict__ bv,
    float* __restrict__ Qp, float* __restrict__ Kp, float* __restrict__ Vp)
{
  const float* X; const float* Wm; const float* b; float* Y;
  if (blockIdx.z == 0)      { X = q; Wm = Wq; b = bq; Y = Qp; }
  else if (blockIdx.z == 1) { X = k; Wm = Wk; b = bk; Y = Kp; }
  else                      { X = v; Wm = Wv; b = bv; Y = Vp; }

  const int lane = threadIdx.x;
  const int m    = lane & 15;
  const int hi   = lane >> 4;
  const int nt   = blockIdx.x;
  const int mt   = blockIdx.y;
  const int arow = mt * 16 + m;
  const int bcol = nt * 16 + m;

  v8f acc = {};
  for (int kk = 0; kk < D_MODEL; kk += 4) {
    const int k0 = kk + 2 * hi;
    v2f a, bb;
    a.x  = Wm[arow * D_MODEL + k0];
    a.y  = Wm[arow * D_MODEL + k0 + 1];
    bb.x = X[(k0)     * NPIX + bcol];
    bb.y = X[(k0 + 1) * NPIX + bcol];
    acc = __builtin_amdgcn_wmma_f32_16x16x4_f32(false, a, false, bb, (short)0, acc, false, false);
  }
#pragma unroll
  for (int r = 0; r < 8; ++r) {
    const int orow = mt * 16 + r + 8 * hi;
    Y[orow * NPIX + nt * 16 + m] = acc[r] + b[orow];
  }
}

// rel GEMM per head: rel[g][kw][p] = sum_d relW[g][kw][d] * Qp[g*32+d][p] + relb[g][kw]
__global__ __launch_bounds__(32) void rel_gemm(
    const float* __restrict__ Qp, const float* __restrict__ relW,
    const float* __restrict__ relb, float* __restrict__ rel)
{
  const int lane = threadIdx.x;
  const int m    = lane & 15;
  const int hi   = lane >> 4;
  const int nt   = blockIdx.x;
  const int mt   = blockIdx.y;
  const int g    = blockIdx.z;
  const int arow = mt * 16 + m;
  const bool aok = (arow < W2);
  const int bcol = nt * 16 + m;

  v8f acc = {};
#pragma unroll
  for (int d = 0; d < HID; d += 4) {
    const int d0 = d + 2 * hi;
    v2f a, bb;
    a.x  = aok ? relW[(g * W2 + arow) * HID + d0]     : 0.0f;
    a.y  = aok ? relW[(g * W2 + arow) * HID + d0 + 1] : 0.0f;
    bb.x = Qp[(g * HID + d0)     * NPIX + bcol];
    bb.y = Qp[(g * HID + d0 + 1) * NPIX + bcol];
    acc = __builtin_amdgcn_wmma_f32_16x16x4_f32(false, a, false, bb, (short)0, acc, false, false);
  }
#pragma unroll
  for (int r = 0; r < 8; ++r) {
    const int kw = mt * 16 + r + 8 * hi;
    if (kw < W2)
      rel[(g * W2 + kw) * NPIX + nt * 16 + m] = acc[r] + relb[g * W2 + kw];
  }
}

// ---------------------------------------------------------------------------
// TDM: DMA one (tdx, tdy, HID) sub-tensor of a (48, 48, 32) fp32 tensor
// (strides 1 / 48 / 2304) into LDS at byte offset lds_off, packed as
// [d][ry][rx] (innermost x). D# bit layout per CDNA5 ISA.
// This toolchain exposes the 6-arg builtin: (v4u, v8i, v4i, v4i, v8i, i32 cpol).
// ---------------------------------------------------------------------------
__device__ __forceinline__ void tdm_load_tile(const float* gsrc, unsigned lds_off,
                                              int tdx, int tdy)
{
  const unsigned long long ga = (unsigned long long)(uintptr_t)gsrc;
  v4u g0;
  g0.x = 1u;                                               // count=1 user descriptor
  g0.y = lds_off;                                          // lds_addr (bytes)
  g0.z = (unsigned)(ga & 0xFFFFFFFFull);                   // global_addr[31:0]
  g0.w = (unsigned)((ga >> 32) & 0x1FFFFFFull) | (2u << 30); // addr[56:32] | type=2

  v8i g1;
  g1[0] = 0x00020000;                    // data_size=2 (4 bytes); no multicast/pad
  g1[1] = (int)(IMG_W << 16);            // tensor_dim0[15:0]=48 in bits 63:48
  g1[2] = (int)(IMG_H << 16);            // tensor_dim0 hi=0 | tensor_dim1[15:0]=48
  g1[3] = (int)(tdx << 16);              // tensor_dim1 hi=0 | tile_dim0
  g1[4] = (int)(tdy | (HID << 16));      // tile_dim1 | tile_dim2=32
  g1[5] = IMG_W;                         // tensor_dim0_stride[31:0] = 48
  g1[6] = (int)(NPIX << 16);             // stride0 hi=0 | tensor_dim1_stride[15:0]=2304
  g1[7] = 0;                             // tensor_dim1_stride hi = 0

  v4i g2;
  g2[0] = HID;                           // tensor_dim2 = 32
  g2[1] = 0;                             // tensor_dim3 (unused, tile_dim3=0)
  g2[2] = 0;                             // tensor_dim2_stride lo (unused)
  g2[3] = 0;                             // stride hi | tile_dim3 = 0

  v4i g3 = {0, 0, 0, 0};                 // dim3 stride / dim4 unused
  v8i gx = {0, 0, 0, 0, 0, 0, 0, 0};     // extra group (unused for <=3D tiles)

  __builtin_amdgcn_tensor_load_to_lds(g0, g1, g2, g3, gx, 0);
}

// Local attention, one (head, 16x8 query tile) per 128-thread block.
// K/V halo staged in LDS by the Tensor Data Mover; two-pass softmax + fused AV.
__global__ __launch_bounds__(128) void attn_tile_kernel(
    const float* __restrict__ Qp, const float* __restrict__ Kp, const float* __restrict__ Vp,
    const float* __restrict__ rel, const float* __restrict__ Vb,
    float* __restrict__ attn, float* __restrict__ Oc, float* __restrict__ invZ)
{
  extern __shared__ float smem[];
  float* Kl = smem;
  float* Vl = smem + KTILE_FLOATS;

  const int g  = blockIdx.z;
  const int x0 = blockIdx.x * TQX;
  const int y0 = blockIdx.y * TQY;

  // halo clamped to the image -> descriptor never goes out of bounds
  const int xs = (x0 - MAX_DIS < 0) ? 0 : x0 - MAX_DIS;
  const int ys = (y0 - MAX_DIS < 0) ? 0 : y0 - MAX_DIS;
  const int xe = (x0 + TQX + MAX_DIS > IMG_W) ? IMG_W : x0 + TQX + MAX_DIS;
  const int ye = (y0 + TQY + MAX_DIS > IMG_H) ? IMG_H : y0 + TQY + MAX_DIS;
  const int tdx = xe - xs;
  const int tdy = ye - ys;
  const int plane = tdx * tdy;

  if (threadIdx.x < 32) {                      // wave 0 drives the TDM
    const float* ksrc = Kp + (size_t)(g * HID) * NPIX + ys * IMG_W + xs;
    const float* vsrc = Vp + (size_t)(g * HID) * NPIX + ys * IMG_W + xs;
    tdm_load_tile(ksrc, (unsigned)(uintptr_t)Kl, tdx, tdy);
    tdm_load_tile(vsrc, (unsigned)(uintptr_t)Vl, tdx, tdy);
    __builtin_amdgcn_s_wait_tensorcnt(0);      // DMA complete before the barrier
  }
  __syncthreads();

  const int tx = threadIdx.x & (TQX - 1);
  const int ty = threadIdx.x >> 4;
  const int x = x0 + tx;
  const int y = y0 + ty;
  const int p = y * IMG_W + x;
  const float Tinv = 0.17677669529663687f;     // 1/sqrt(HID)

  float qv[HID];
#pragma unroll
  for (int d = 0; d < HID; ++d) qv[d] = Qp[(g * HID + d) * NPIX + p] * Tinv;

  // pass 1: running max over the valid window
  float mx = -1e30f;
  for (int i = 0; i < WSZ; ++i) {
    const int yy = y + i - MAX_DIS;
    if (yy < 0 || yy >= IMG_H) continue;
    const int ry = yy - ys;
    for (int j = 0; j < WSZ; ++j) {
      const int xx = x + j - MAX_DIS;
      if (xx < 0 || xx >= IMG_W) continue;
      const int kw   = i * WSZ + j;
      const int base = ry * tdx + (xx - xs);
      float s = rel[(g * W2 + kw) * NPIX + p];
#pragma unroll
      for (int d = 0; d < HID; ++d) s += qv[d] * Kl[base + d * plane];
      mx = fmaxf(mx, s);
    }
  }

  // pass 2: exp, partition sum, fused AV (+V_bias)
  float Z = 0.0f;
  float Ov[HID];
#pragma unroll
  for (int d = 0; d < HID; ++d) Ov[d] = 0.0f;

  for (int i = 0; i < WSZ; ++i) {
    const int yy = y + i - MAX_DIS;
    const bool yok = (yy >= 0 && yy < IMG_H);
    const int ry = yy - ys;
    for (int j = 0; j < WSZ; ++j) {
      const int xx = x + j - MAX_DIS;
      const int kw = i * WSZ + j;
      const bool ok = yok && (xx >= 0) && (xx < IMG_W);
      float e = 0.0f;
      if (ok) {
        const int base = ry * tdx + (xx - xs);
        float s = rel[(g * W2 + kw) * NPIX + p];
#pragma unroll
        for (int d = 0; d < HID; ++d) s += qv[d] * Kl[base + d * plane];
        e = __expf(s - mx);
        Z += e;
#pragma unroll
        for (int d = 0; d < HID; ++d)
          Ov[d] += e * (Vl[base + d * plane] + Vb[(g * HID + d) * W2 + kw]);
      }
      // reference masks OOB to -1e8 -> exp underflows to exactly 0 in fp32
      attn[(g * W2 + kw) * NPIX + p] = e;
    }
  }
  const float iz = 1.0f / Z;                   // center tap always valid -> Z > 0
  invZ[g * NPIX + p] = iz;
#pragma unroll
  for (int d = 0; d < HID; ++d) Oc[(g * HID + d) * NPIX + p] = Ov[d] * iz;
}

// attn[g][kw][p] *= 1/Z[g][p]
__global__ __launch_bounds__(256) void attn_scale(
    float* __restrict__ attn, const float* __restrict__ invZ)
{
  const int idx = blockIdx.x * blockDim.x + threadIdx.x;
  if (idx >= NHEAD * W2 * NPIX) return;
  const int p = idx % NPIX;
  const int g = idx / (W2 * NPIX);
  attn[idx] *= invZ[g * NPIX + p];
}

// fc GEMM: out[p][o] = sum_c fcW[o][c] * Oc[c][p] + fcb[o]  (stored (p,256) row-major)
__global__ __launch_bounds__(32) void fc_gemm(
    const float* __restrict__ Oc, const float* __restrict__ fcW,
    const float* __restrict__ fcb, float* __restrict__ out)
{
  const int lane = threadIdx.x;
  const int m    = lane & 15;
  const int hi   = lane >> 4;
  const int nt   = blockIdx.x;
  const int mt   = blockIdx.y;
  const int arow = mt * 16 + m;
  const int bcol = nt * 16 + m;

  v8f acc = {};
  for (int kk = 0; kk < D_MODEL; kk += 4) {
    const int k0 = kk + 2 * hi;
    v2f a, bb;
    a.x  = fcW[arow * D_MODEL + k0];
    a.y  = fcW[arow * D_MODEL + k0 + 1];
    bb.x = Oc[(k0)     * NPIX + bcol];
    bb.y = Oc[(k0 + 1) * NPIX + bcol];
    acc = __builtin_amdgcn_wmma_f32_16x16x4_f32(false, a, false, bb, (short)0, acc, false, false);
  }
#pragma unroll
  for (int r = 0; r < 8; ++r) {
    const int o  = mt * 16 + r + 8 * hi;
    const int pp = nt * 16 + m;
    out[pp * D_MODEL + o] = acc[r] + fcb[o];
  }
}

extern "C" void kernel_launch(void* const* d_in, const int* in_sizes, int n_in,
                              void* d_out, int out_size, void* d_ws, size_t ws_size,
                              hipStream_t stream)
{
  (void)in_sizes; (void)n_in; (void)out_size; (void)ws_size;

  const float* q    = (const float*)d_in[0];
  const float* k    = (const float*)d_in[1];
  const float* v    = (const float*)d_in[2];
  const float* Wq   = (const float*)d_in[3];
  const float* bq   = (const float*)d_in[4];
  const float* Wk   = (const float*)d_in[5];
  const float* bk   = (const float*)d_in[6];
  const float* Wv   = (const float*)d_in[7];
  const float* bv   = (const float*)d_in[8];
  const float* relW = (const float*)d_in[9];
  const float* relb = (const float*)d_in[10];
  const float* Vb   = (const float*)d_in[11];
  const float* fcW  = (const float*)d_in[12];
  const float* fcb  = (const float*)d_in[13];

  float* out  = (float*)d_out;               // (2304, 1, 256)
  float* attn = out + NPIX * D_MODEL;        // (1, 8, 225, 2304)

  // workspace layout (floats): ~26.1 MB
  float* ws   = (float*)d_ws;
  float* Qp   = ws;                          // 256*2304
  float* Kp   = Qp  + D_MODEL * NPIX;        // 256*2304
  float* Vp   = Kp  + D_MODEL * NPIX;        // 256*2304
  float* rel  = Vp  + D_MODEL * NPIX;        // 8*225*2304
  float* Oc   = rel + NHEAD * W2 * NPIX;     // 256*2304
  float* invZ = Oc  + D_MODEL * NPIX;        // 8*2304

  proj3_gemm<<<dim3(NPIX / 16, D_MODEL / 16, 3), 32, 0, stream>>>(
      q, k, v, Wq, Wk, Wv, bq, bk, bv, Qp, Kp, Vp);

  rel_gemm<<<dim3(NPIX / 16, (W2 + 15) / 16, NHEAD), 32, 0, stream>>>(
      Qp, relW, relb, rel);

  attn_tile_kernel<<<dim3(IMG_W / TQX, IMG_H / TQY, NHEAD), 128,
                     2 * KTILE_FLOATS * sizeof(float), stream>>>(
      Qp, Kp, Vp, rel, Vb, attn, Oc, invZ);

  attn_scale<<<(NHEAD * W2 * NPIX + 255) / 256, 256, 0, stream>>>(attn, invZ);

  fc_gemm<<<dim3(NPIX / 16, D_MODEL / 16), 32, 0, stream>>>(Oc, fcW, fcb, out);
}